// PolicyNetGCN_3822520893862
// MI455X (gfx1250) — compile-verified
//
#include <hip/hip_runtime.h>
#include <hip/hip_bf16.h>

typedef float v2f __attribute__((ext_vector_type(2)));
typedef float v8f __attribute__((ext_vector_type(8)));

#define GCN_D      64
#define GCN_B      4
#define FLAT       (GCN_D * GCN_B)      // 256
#define ROWS_PER_BLOCK 128              // 8 waves * 16 rows
#define THREADS    256
#define EDGES_PER_BLOCK 4
#define XS         68                   // padded LDS row stride (floats): bank-conflict-free

// ---------------------------------------------------------------------------
// 1) Transpose state [B, N, D] -> X [N, B, D]  (row layout for spmm: [N, 256])
// ---------------------------------------------------------------------------
__global__ void gcn_transpose_state(const float* __restrict__ state,
                                    float* __restrict__ X, int n_nodes) {
    long long total = (long long)n_nodes * FLAT;
    long long i = (long long)blockIdx.x * blockDim.x + threadIdx.x;
    if (i >= total) return;
    int n   = (int)(i >> 8);        // / 256
    int rem = (int)(i & 255);
    int b   = rem >> 6;
    int d   = rem & 63;
    X[i] = state[((long long)b * n_nodes + n) * GCN_D + d];
}

// ---------------------------------------------------------------------------
// 2) Dense layer: S[row, :] = act(X[row, :]) * W + bias, rows = n*B+b (200k)
//    One wave -> 16x64 output strip via V_WMMA_F32_16X16X4_F32.
//    W kept transposed+padded in LDS so every A/B operand is one ds_load_b64
//    hitting all 64 banks exactly once. Output tile staged through LDS for
//    coalesced float4 stores.
// ---------------------------------------------------------------------------
__global__ void gcn_gemm_wmma(const float* __restrict__ X,
                              const float* __restrict__ W,
                              const float* __restrict__ bias,
                              float* __restrict__ S,
                              int nrows, int relu_on_load) {
    __shared__ __align__(16) float ldsWt[GCN_D * XS];           // 17.4 KB, Wt[col][k]
    __shared__ __align__(16) float ldsX[ROWS_PER_BLOCK * XS];   // 34.8 KB, X[r][k] padded

    const int t = threadIdx.x;

    // Stage W transposed: ldsWt[col*XS + k] = W[k*64 + col] (coalesced global reads).
    #pragma unroll
    for (int i = t; i < GCN_D * GCN_D; i += THREADS) {
        const int k = i >> 6, col = i & 63;
        ldsWt[col * XS + k] = W[i];
    }

    // Stage a 128-row strip of X (padded rows); fuse ReLU of previous layer.
    const int row0 = blockIdx.x * ROWS_PER_BLOCK;
    for (int i = t; i < ROWS_PER_BLOCK * GCN_D; i += THREADS) {
        const int rl = i >> 6, c = i & 63;
        const int r = row0 + rl;
        float v = (r < nrows) ? X[(long long)r * GCN_D + c] : 0.0f;
        if (relu_on_load) v = fmaxf(v, 0.0f);
        ldsX[rl * XS + c] = v;
    }
    __syncthreads();

    const int wave  = t >> 5;        // 0..7 -> which 16-row strip
    const int lane  = t & 31;
    const int l15   = lane & 15;     // A: row-in-tile / B: col-in-tile / D: N
    const int khalf = lane >> 4;     // selects K pair {0,1} or {2,3}
    const int wrow0 = wave * 16;

    v8f acc[4];
    #pragma unroll
    for (int j = 0; j < 4; ++j) acc[j] = (v8f){0.f,0.f,0.f,0.f,0.f,0.f,0.f,0.f};

    #pragma unroll
    for (int kk = 0; kk < 16; ++kk) {
        const int k = kk * 4 + khalf * 2;
        // A 16x4 fp32: lanes 0-15 hold K=0,1 of row M=lane; lanes 16-31 hold K=2,3.
        const v2f a = *(const v2f*)(ldsX + (wrow0 + l15) * XS + k);
        #pragma unroll
        for (int j = 0; j < 4; ++j) {
            // B 4x16 fp32 from transposed W: contiguous pair -> ds_load_b64.
            const v2f b = *(const v2f*)(ldsWt + (j * 16 + l15) * XS + k);
            acc[j] = __builtin_amdgcn_wmma_f32_16x16x4_f32(
                /*neg_a=*/false, a, /*neg_b=*/false, b,
                /*c_mod=*/(short)0, acc[j], /*reuse_a=*/false, /*reuse_b=*/false);
        }
    }

    // Epilogue: stage D tiles (+bias) into reused ldsX, then coalesced b128 stores.
    __syncthreads();
    const int mbase = (lane >> 4) ? 8 : 0;
    #pragma unroll
    for (int j = 0; j < 4; ++j) {
        const int colN = j * 16 + l15;
        const float bc = bias[colN];
        #pragma unroll
        for (int v = 0; v < 8; ++v) {
            const int rl = wrow0 + mbase + v;
            ldsX[rl * XS + colN] = acc[j][v] + bc;
        }
    }
    __syncthreads();
    for (int q = t; q < ROWS_PER_BLOCK * GCN_D / 4; q += THREADS) {
        const int idx = q * 4;
        const int rl = idx >> 6, c = idx & 63;
        const int r = row0 + rl;
        if (r < nrows) {
            const float4 val = *(const float4*)(ldsX + rl * XS + c);
            *(float4*)(S + (long long)r * GCN_D + c) = val;
        }
    }
}

// ---------------------------------------------------------------------------
// 3) SpMM: O[row_e, :] += w_e * S[col_e, :]   (256-wide rows, L2-resident)
// ---------------------------------------------------------------------------
__global__ void gcn_spmm_atomic(const float* __restrict__ S,
                                const int* __restrict__ erow,
                                const int* __restrict__ ecol,
                                const float* __restrict__ ew,
                                float* __restrict__ O, int n_edges) {
    const int c = threadIdx.x;  // 0..255 column within flat row
    const int e0 = blockIdx.x * EDGES_PER_BLOCK;
    #pragma unroll
    for (int i = 0; i < EDGES_PER_BLOCK; ++i) {
        const int e = e0 + i;
        if (e >= n_edges) return;
        const int src = ecol[e];
        const int dst = erow[e];
        const float w = ew[e];
        const float v = w * S[(long long)src * FLAT + c];
        atomicAdd(&O[(long long)dst * FLAT + c], v);
    }
}

// ---------------------------------------------------------------------------
// 4) Head: out[b*N + n] = sum_d relu(X[n, b*64 + d]) * w_out[d]
//    One wave per (n, b); 8 waves/block -> 2 nodes per block.
// ---------------------------------------------------------------------------
__global__ void gcn_head(const float* __restrict__ X,
                         const float* __restrict__ wout,
                         float* __restrict__ out, int n_nodes) {
    const int t = threadIdx.x;
    const int wave = t >> 5;
    const int lane = t & 31;
    const int n = blockIdx.x * 2 + (wave >> 2);
    const int b = wave & 3;
    if (n >= n_nodes) return;
    const float* xp = X + (long long)n * FLAT + b * GCN_D;
    float s = fmaxf(xp[lane], 0.0f) * wout[lane]
            + fmaxf(xp[lane + 32], 0.0f) * wout[lane + 32];
    #pragma unroll
    for (int off = 16; off > 0; off >>= 1) s += __shfl_down(s, off, 32);
    if (lane == 0) out[(long long)b * n_nodes + n] = s;
}

// ---------------------------------------------------------------------------
extern "C" void kernel_launch(void* const* d_in, const int* in_sizes, int n_in,
                              void* d_out, int out_size, void* d_ws, size_t ws_size,
                              hipStream_t stream) {
    const float* state = (const float*)d_in[0];
    const int*   erow  = (const int*)d_in[1];
    const int*   ecol  = (const int*)d_in[2];
    const float* ew    = (const float*)d_in[3];
    const float* W1    = (const float*)d_in[4];
    const float* b1    = (const float*)d_in[5];
    const float* W2    = (const float*)d_in[6];
    const float* b2    = (const float*)d_in[7];
    const float* wout  = (const float*)d_in[8];

    const int n_nodes = in_sizes[0] / (GCN_B * GCN_D);  // 50000
    const int n_edges = in_sizes[1];                    // 800000
    const int nrows   = n_nodes * GCN_B;                // 200000

    const size_t bufBytes = (size_t)nrows * GCN_D * sizeof(float);  // 51.2 MB
    const size_t align    = 256;
    const size_t bufStride = (bufBytes + align - 1) / align * align;
    float* bufA = (float*)((char*)d_ws + 0 * bufStride);
    float* bufB = (float*)((char*)d_ws + 1 * bufStride);
    float* bufC = (float*)((char*)d_ws + 2 * bufStride);

    const long long totalElems = (long long)nrows * GCN_D;
    const int tGrid = (int)((totalElems + THREADS - 1) / THREADS);
    const int gGrid = (nrows + ROWS_PER_BLOCK - 1) / ROWS_PER_BLOCK;
    const int eGrid = (n_edges + EDGES_PER_BLOCK - 1) / EDGES_PER_BLOCK;
    const int hGrid = (n_nodes + 1) / 2;

    // ---- layer 1 ----
    gcn_transpose_state<<<tGrid, THREADS, 0, stream>>>(state, bufA, n_nodes);
    gcn_gemm_wmma<<<gGrid, THREADS, 0, stream>>>(bufA, W1, b1, bufB, nrows, 0);
    hipMemsetAsync(bufC, 0, bufBytes, stream);
    gcn_spmm_atomic<<<eGrid, THREADS, 0, stream>>>(bufB, erow, ecol, ew, bufC, n_edges);

    // ---- layer 2 (ReLU of layer-1 output fused into GEMM load) ----
    gcn_gemm_wmma<<<gGrid, THREADS, 0, stream>>>(bufC, W2, b2, bufB, nrows, 1);
    hipMemsetAsync(bufA, 0, bufBytes, stream);
    gcn_spmm_atomic<<<eGrid, THREADS, 0, stream>>>(bufB, erow, ecol, ew, bufA, n_edges);

    // ---- head (ReLU fused) ----
    gcn_head<<<hGrid, THREADS, 0, stream>>>(bufA, wout, (float*)d_out, n_nodes);
}